// StructuredStateSpace_47596827574495
// MI455X (gfx1250) — compile-verified
//
#include <hip/hip_runtime.h>
#include <hip/hip_bf16.h>

typedef __attribute__((ext_vector_type(2))) float v2f;
typedef __attribute__((ext_vector_type(8))) float v8f;
typedef __attribute__((ext_vector_type(4))) float f4;

#define B_   32
#define S_   2048
#define DM_  1024
#define N_   256
#define LN_EPS 1e-5f

// ---------------------------------------------------------------------------
// Kernel 1: u[row] = mean over DM_ of x[row, :], via V_WMMA_F32_16X16X4_F32.
// A = ones(16x4)  =>  D[m,n] = C[m,n] + sum_k B[k,n]  (column sums, exact f32).
// Each wave reduces 16 rows; lane l (and l+16) feed column l&15.
// Lanes 0..15 load float4 at col j, lanes 16..31 at col j+4 => 8 cols / iter,
// 2 WMMAs / iter. Result for row (rowbase + n) sits in D vgpr0, lane n (M=0).
// ---------------------------------------------------------------------------
__global__ __launch_bounds__(256) void k_rowmean(const float* __restrict__ x,
                                                 float* __restrict__ u) {
  const int lane    = threadIdx.x & 31;
  const int wid     = threadIdx.x >> 5;
  const int rowbase = blockIdx.x * 128 + wid * 16;   // 8 waves * 16 rows
  const int r       = rowbase + (lane & 15);
  const int half    = lane >> 4;                     // 0 or 1
  const float* rp   = x + (size_t)r * DM_ + half * 4;

  v2f ones; ones.x = 1.0f; ones.y = 1.0f;
  v8f acc = {};
  #pragma unroll 4
  for (int j = 0; j < DM_; j += 8) {
    f4 v = *(const f4*)(rp + j);
    v2f b0; b0.x = v.x; b0.y = v.y;
    v2f b1; b1.x = v.z; b1.y = v.w;
    acc = __builtin_amdgcn_wmma_f32_16x16x4_f32(false, ones, false, b0,
                                                (short)0, acc, false, false);
    acc = __builtin_amdgcn_wmma_f32_16x16x4_f32(false, ones, false, b1,
                                                (short)0, acc, false, false);
  }
  if (lane < 16) u[rowbase + lane] = acc[0] * (1.0f / (float)DM_);
}

// ---------------------------------------------------------------------------
// Kernel 2: sequential scan, one wave32 per batch. Lane owns 8 of 256 states.
// h = a*h + b*u_t ; h = LN(h)*gamma+beta ; s = <h, c>. All reductions are
// intra-wave __shfl_xor trees (no barriers, minimal dependent-chain latency).
// ---------------------------------------------------------------------------
__global__ __launch_bounds__(32) void k_scan(const float* __restrict__ u,
                                             const float* __restrict__ log_lambda,
                                             const float* __restrict__ log_b,
                                             const float* __restrict__ cvec,
                                             const float* __restrict__ log_step,
                                             const float* __restrict__ gamma,
                                             const float* __restrict__ beta,
                                             float* __restrict__ sout) {
  const int lane = threadIdx.x;
  const int b    = blockIdx.x;
  const float step = expf(log_step[0]);

  float ad[8], bd[8], g[8], be[8], cw[8], h[8];
  #pragma unroll
  for (int j = 0; j < 8; ++j) {
    int n = lane * 8 + j;
    float lam = -expf(log_lambda[n]);
    float a   = (2.0f + step * lam) / (2.0f - step * lam);
    ad[j] = a;
    bd[j] = step * (1.0f + a) * expf(log_b[n]) * 0.5f;
    g[j]  = gamma[n];
    be[j] = beta[n];
    cw[j] = cvec[n];
    h[j]  = 0.0f;
  }

  const float* ub = u    + (size_t)b * S_;
  float*       sb = sout + (size_t)b * S_;

  float ucur = ub[lane];                               // 32 u values per chunk
  for (int t0 = 0; t0 < S_; t0 += 32) {
    float unext = (t0 + 32 < S_) ? ub[t0 + 32 + lane] : 0.0f;  // prefetch
    for (int i = 0; i < 32; ++i) {
      float ut = __shfl(ucur, i, 32);
      float sum = 0.0f, sq = 0.0f;
      #pragma unroll
      for (int j = 0; j < 8; ++j) {
        h[j] = fmaf(ad[j], h[j], bd[j] * ut);
        sum += h[j];
        sq   = fmaf(h[j], h[j], sq);
      }
      #pragma unroll
      for (int off = 16; off > 0; off >>= 1) {
        sum += __shfl_xor(sum, off, 32);
        sq  += __shfl_xor(sq,  off, 32);
      }
      float mu   = sum * (1.0f / (float)N_);
      float var  = sq  * (1.0f / (float)N_) - mu * mu;
      float rstd = rsqrtf(var + LN_EPS);
      float sdot = 0.0f;
      #pragma unroll
      for (int j = 0; j < 8; ++j) {
        float hn = fmaf((h[j] - mu) * rstd, g[j], be[j]);
        h[j] = hn;
        sdot = fmaf(hn, cw[j], sdot);
      }
      #pragma unroll
      for (int off = 16; off > 0; off >>= 1) sdot += __shfl_xor(sdot, off, 32);
      if (lane == 0) sb[t0 + i] = sdot;                // fire-and-forget store
    }
    ucur = unext;
  }
}

// ---------------------------------------------------------------------------
// Kernel 3: out = k1*x + k2*s[row], streaming float4 (bandwidth-bound).
// k1 = sig(alpha) + (1-sig(alpha))*exp(log_d) ; k2 = 1-sig(alpha)
// ---------------------------------------------------------------------------
__global__ __launch_bounds__(256) void k_out(const float* __restrict__ x,
                                             const float* __restrict__ s,
                                             const float* __restrict__ log_d,
                                             const float* __restrict__ alpha,
                                             float* __restrict__ out) {
  size_t i = (size_t)blockIdx.x * blockDim.x + threadIdx.x;   // float4 index
  const float asig = 1.0f / (1.0f + expf(-alpha[0]));
  const float dval = expf(log_d[0]);
  const float k2   = 1.0f - asig;
  const float k1   = asig + k2 * dval;

  size_t row = i >> 8;              // 1024/4 = 256 float4 per row
  float sv   = k2 * s[row];
  f4 xv = ((const f4*)x)[i];
  f4 o;
  o.x = fmaf(k1, xv.x, sv);
  o.y = fmaf(k1, xv.y, sv);
  o.z = fmaf(k1, xv.z, sv);
  o.w = fmaf(k1, xv.w, sv);
  ((f4*)out)[i] = o;
}

extern "C" void kernel_launch(void* const* d_in, const int* in_sizes, int n_in,
                              void* d_out, int out_size, void* d_ws, size_t ws_size,
                              hipStream_t stream) {
  const float* x          = (const float*)d_in[0];   // [B,S,DM]
  const float* log_lambda = (const float*)d_in[1];   // [N]
  const float* log_b      = (const float*)d_in[2];   // [N,1]
  const float* cvec       = (const float*)d_in[3];   // [1,N]
  const float* log_d      = (const float*)d_in[4];   // [1]
  const float* log_step   = (const float*)d_in[5];   // [1]
  const float* alpha      = (const float*)d_in[6];   // [1]
  const float* ln_gamma   = (const float*)d_in[7];   // [N]
  const float* ln_beta    = (const float*)d_in[8];   // [N]
  float* out = (float*)d_out;

  float* u = (float*)d_ws;               // B*S floats
  float* s = u + (size_t)B_ * S_;        // B*S floats

  // 1) u = row means of x  (65536 rows, 128 rows per 256-thread block)
  k_rowmean<<<(B_ * S_) / 128, 256, 0, stream>>>(x, u);

  // 2) sequential scan, one wave per batch
  k_scan<<<B_, 32, 0, stream>>>(u, log_lambda, log_b, cvec, log_step,
                                ln_gamma, ln_beta, s);

  // 3) streaming output blend
  const size_t n4 = (size_t)B_ * S_ * DM_ / 4;
  k_out<<<(unsigned)(n4 / 256), 256, 0, stream>>>(x, s, log_d, alpha, out);
}